// FeaturesLoss_45028437131690
// MI455X (gfx1250) — compile-verified
//
#include <hip/hip_runtime.h>

// ---------------------------------------------------------------------------
// FeaturesLoss for MI455X (gfx1250, wave32, WMMA)
//   loss = 0.5 * [ sum_{same class} d2/eq + sum_{diff class} relu(1-d)^2/neq ]
//   d2 = ||f_i||^2 + ||f_j||^2 - 2 * (F F^T)_ij   (clamped >= 0)
// Gram matrix via V_WMMA_F32_16X16X4_F32. F (8MB) lives in L2 (192MB), so the
// kernel is fp32-matrix-pipe bound; staging uses GLOBAL_LOAD_ASYNC_TO_LDS_B128
// (ASYNCcnt) with double-buffered LDS panels to hide load latency.
// ---------------------------------------------------------------------------

typedef __attribute__((ext_vector_type(2))) float v2f;
typedef __attribute__((ext_vector_type(8))) float v8f;

#define NPTS 4096
#define DDIM 512
#define NCLS 100
#define TILE 128
#define KT   32
#define NKT  (DDIM / KT)   // 16 K-panels
#define LSTR 36            // LDS row stride floats: 144B (16B aligned, bank-friendly)
#define PANEL (TILE * LSTR)

__global__ __launch_bounds__(32) void init_ws_kernel(float* wsf, int* eqp) {
    if (threadIdx.x == 0) {
        wsf[0] = 0.0f;   // S_pos
        wsf[1] = 0.0f;   // S_neg
        *eqp   = 0;
    }
}

// One wave (32 lanes) per row: sq[row] = sum_k f[row][k]^2
__global__ __launch_bounds__(256) void row_sq_kernel(const float* __restrict__ f,
                                                     float* __restrict__ sq) {
    const int wave = threadIdx.x >> 5;
    const int lane = threadIdx.x & 31;
    const int row  = blockIdx.x * 8 + wave;
    const float* p = f + (size_t)row * DDIM;
    float s = 0.0f;
#pragma unroll
    for (int k = 0; k < DDIM / 32; ++k) {
        float v = p[lane + 32 * k];
        s += v * v;
    }
#pragma unroll
    for (int m = 16; m >= 1; m >>= 1) s += __shfl_xor(s, m, 32);
    if (lane == 0) sq[row] = s;
}

// eq_sum = sum_k count_k^2 via a tiny LDS histogram (y in [0,100))
__global__ __launch_bounds__(256) void class_hist_kernel(const int* __restrict__ y,
                                                         int* __restrict__ eqp) {
    __shared__ int cnt[NCLS];
    for (int i = threadIdx.x; i < NCLS; i += blockDim.x) cnt[i] = 0;
    __syncthreads();
    for (int i = threadIdx.x; i < NPTS; i += blockDim.x) atomicAdd(&cnt[y[i]], 1);
    __syncthreads();
    if (threadIdx.x == 0) {
        long long e = 0;
        for (int k = 0; k < NCLS; ++k) e += (long long)cnt[k] * (long long)cnt[k];
        *eqp = (int)e;
    }
}

// Low 32 bits of a flat shared pointer are the workgroup-relative LDS address
// (ISA: flat->LDS maps via addr[31:0] truncation).
__device__ __forceinline__ uint32_t lds_addr32(const void* p) {
    return (uint32_t)(uintptr_t)p;
}

// Async memory->LDS, 16B per lane, no data VGPRs; tracked with ASYNCcnt.
__device__ __forceinline__ void async_load_b128(uint32_t lds_off, const float* gptr) {
    asm volatile("global_load_async_to_lds_b128 %0, %1, off"
                 :: "v"(lds_off), "v"(gptr)
                 : "memory");
}

// Issue one 128x32 A-panel + one 128x32 B-panel into the given LDS buffer.
// 8 async b128 ops per thread per tile (=> per-wave ASYNCcnt +8).
__device__ __forceinline__ void issue_panels(const float* __restrict__ f,
                                             uint32_t ldsA, uint32_t ldsB,
                                             int ibase, int jbase, int k0, int tid) {
#pragma unroll
    for (int r = 0; r < 4; ++r) {
        int idx = tid + 256 * r;           // 0..1023
        int row = idx >> 3;                // 0..127
        int c4  = (idx & 7) << 2;          // 0,4,...,28
        uint32_t off = (uint32_t)(row * LSTR + c4) * 4u;
        const float* ga = f + (size_t)(ibase + row) * DDIM + k0 + c4;
        const float* gb = f + (size_t)(jbase + row) * DDIM + k0 + c4;
        async_load_b128(ldsA + off, ga);
        async_load_b128(ldsB + off, gb);
    }
}

// 128x128 gram tile per block (8 waves), double-buffered async staging,
// fused distance + masked-hinge epilogue.
__global__ __launch_bounds__(256) void gram_loss_kernel(const float* __restrict__ f,
                                                        const float* __restrict__ sq,
                                                        const int*   __restrict__ y,
                                                        float*       __restrict__ accums) {
    __shared__ __align__(16) float lds[4 * PANEL];   // [buf(2)][A,B][128][LSTR]
    __shared__ float red[2];

    const int tid  = threadIdx.x;
    const int wave = tid >> 5;
    const int lane = tid & 31;
    const int hi   = lane >> 4;   // half-wave select (K pair / M group per ISA layout)
    const int lo   = lane & 15;

    const int ibase = blockIdx.y * TILE;   // rows of this tile (A panel)
    const int jbase = blockIdx.x * TILE;   // cols of this tile (B panel)

    const uint32_t ldsA0 = lds_addr32(&lds[0 * PANEL]);
    const uint32_t ldsB0 = lds_addr32(&lds[1 * PANEL]);
    const uint32_t ldsA1 = lds_addr32(&lds[2 * PANEL]);
    const uint32_t ldsB1 = lds_addr32(&lds[3 * PANEL]);

    v8f acc[8] = {};   // wave's 16 rows x 128 cols, 8 sub-tiles of 16x16

    // Prologue: async-stage K-panel 0 into buffer 0.
    issue_panels(f, ldsA0, ldsB0, ibase, jbase, 0, tid);

    for (int kt = 0; kt < NKT; ++kt) {
        const int cur = kt & 1;
        if (kt + 1 < NKT) {
            // Prefetch next K-panel into the other buffer, then wait until only
            // those 8 in-flight ops remain => current panel is resident.
            issue_panels(f,
                         cur ? ldsA0 : ldsA1,
                         cur ? ldsB0 : ldsB1,
                         ibase, jbase, (kt + 1) * KT, tid);
            asm volatile("s_wait_asynccnt 0x8" ::: "memory");
        } else {
            asm volatile("s_wait_asynccnt 0x0" ::: "memory");
        }
        __syncthreads();

        const float* pA = &lds[(cur ? 2 : 0) * PANEL];
        const float* pB = &lds[(cur ? 3 : 1) * PANEL];

        // K-loop: 8 WMMAs per K-step of 4.
        // A 16x4 layout: lane<16 holds (K=k,k+1) of M=lo; lane>=16 holds (K=k+2,k+3).
        // B 4x16 layout is the transpose-symmetric striping -> same LDS indexing.
#pragma unroll
        for (int kk = 0; kk < KT; kk += 4) {
            v2f a = *(const v2f*)(&pA[(wave * 16 + lo) * LSTR + kk + 2 * hi]);
#pragma unroll
            for (int t = 0; t < 8; ++t) {
                v2f b = *(const v2f*)(&pB[(t * 16 + lo) * LSTR + kk + 2 * hi]);
                acc[t] = __builtin_amdgcn_wmma_f32_16x16x4_f32(
                    /*neg_a=*/false, a, /*neg_b=*/false, b,
                    /*c_mod=*/(short)0, acc[t],
                    /*reuse_a=*/false, /*reuse_b=*/false);
            }
        }
        __syncthreads();   // all waves done reading before this buffer is refilled
    }

    // Epilogue. C/D layout: VGPR v, lane l -> row M = v + 8*(l>=16), col N = l&15.
    float sqi[8];
    int   yi[8];
#pragma unroll
    for (int v = 0; v < 8; ++v) {
        int i = ibase + wave * 16 + v + 8 * hi;
        sqi[v] = sq[i];
        yi[v]  = y[i];
    }

    float pos = 0.0f, neg = 0.0f;
#pragma unroll
    for (int t = 0; t < 8; ++t) {
        int   j   = jbase + t * 16 + lo;
        float sqj = sq[j];
        int   yj  = y[j];
#pragma unroll
        for (int v = 0; v < 8; ++v) {
            float g  = acc[t][v];
            float d2 = fmaxf(sqi[v] + sqj - 2.0f * g, 0.0f);
            if (yi[v] == yj) {
                pos += d2;
            } else {
                float dn = sqrtf(d2 + 1e-15f);
                float m  = fmaxf(1.0f - dn, 0.0f);   // MARGIN = 1.0
                neg += m * m;
            }
        }
    }

    // Block reduction: wave shuffle -> LDS atomics -> one global atomic pair.
    if (tid == 0) { red[0] = 0.0f; red[1] = 0.0f; }
    __syncthreads();
#pragma unroll
    for (int m = 16; m >= 1; m >>= 1) {
        pos += __shfl_xor(pos, m, 32);
        neg += __shfl_xor(neg, m, 32);
    }
    if (lane == 0) {
        atomicAdd(&red[0], pos);
        atomicAdd(&red[1], neg);
    }
    __syncthreads();
    if (tid == 0) {
        atomicAdd(&accums[0], red[0]);
        atomicAdd(&accums[1], red[1]);
    }
}

__global__ __launch_bounds__(32) void finalize_kernel(const float* __restrict__ accums,
                                                      const int*   __restrict__ eqp,
                                                      float*       __restrict__ out) {
    if (threadIdx.x == 0) {
        float eq  = (float)(*eqp);
        float neq = (float)NPTS * (float)NPTS - eq;
        out[0] = 0.5f * (accums[0] / eq + accums[1] / neq);
    }
}

extern "C" void kernel_launch(void* const* d_in, const int* in_sizes, int n_in,
                              void* d_out, int out_size, void* d_ws, size_t ws_size,
                              hipStream_t stream) {
    const int*   y = (const int*)d_in[0];    // y_true, int32, [4096]
    const float* f = (const float*)d_in[1];  // f_pred, float32, [4096,512]

    float* wsf = (float*)d_ws;     // [0]=S_pos, [1]=S_neg
    int*   eqp = (int*)(wsf + 2);  // eq_sum
    float* sq  = wsf + 16;         // [4096] row squared norms

    init_ws_kernel<<<1, 32, 0, stream>>>(wsf, eqp);
    row_sq_kernel<<<NPTS / 8, 256, 0, stream>>>(f, sq);
    class_hist_kernel<<<1, 256, 0, stream>>>(y, eqp);

    dim3 grid(NPTS / TILE, NPTS / TILE);   // 32 x 32 blocks
    gram_loss_kernel<<<grid, 256, 0, stream>>>(f, sq, y, wsf);

    finalize_kernel<<<1, 32, 0, stream>>>(wsf, eqp, (float*)d_out);
}